// DotAttentionLayer_24532853195442
// MI455X (gfx1250) — compile-verified
//
#include <hip/hip_runtime.h>

// DotAttentionLayer on MI455X (gfx1250, wave32).
// Per batch b: scores = outer(q_b, k_b)/64 (rank-1), softmax over j, out = dist @ v.
// d_out = [ output (16*4096 f32) | distribution (16*4096*4096 f32, ~1.07 GB) ].
// Store-bandwidth bound: 1.07 GB / 23.3 TB/s ~= 46 us floor. Inputs L2-resident.
// Row sums l_i and acc_i = sum_j e_ij * v_j are fused into one chained
// V_WMMA_F32_16X16X4_F32 stream using B = [ones | v | 0...].
// Distribution written with non-temporal b128 stores (write-once stream,
// 5x larger than L2 -> do not allocate).

typedef __attribute__((ext_vector_type(2))) float v2f;
typedef __attribute__((ext_vector_type(4))) float v4f;
typedef __attribute__((ext_vector_type(8))) float v8f;

#define B_DIM 16
#define N_DIM 4096
#define TPB   256
#define WAVES 8                       // 256 threads / wave32
#define JSTRIPE (N_DIM / WAVES)       // 512 columns per wave
#define ROWS  16                      // softmax rows per block

__launch_bounds__(TPB, 2)
__global__ void attn_rank1_softmax(const float* __restrict__ q,
                                   const float* __restrict__ v,
                                   const float* __restrict__ k,
                                   float* __restrict__ out,     // [B,N]
                                   float* __restrict__ dist) {  // [B,N,N]
    __shared__ float sk[N_DIM];        // 16 KB: k row of this batch
    __shared__ float sv[N_DIM];        // 16 KB: v row of this batch
    __shared__ float red[TPB];
    __shared__ float sc1[ROWS];        // q_i * log2(e)/64
    __shared__ float sm2[ROWS];        // row max in log2 domain
    __shared__ float slinv[ROWS];      // 1 / l_i
    __shared__ float sacc[ROWS];       // sum_j e_ij * v_j
    __shared__ float lsum[WAVES][ROWS];
    __shared__ float lacc[WAVES][ROWS];
    __shared__ float skmax, skmin;

    const int tid = threadIdx.x;
    const int b   = blockIdx.x >> 8;          // 256 row-tiles per batch
    const int i0  = (blockIdx.x & 255) << 4;  // first of 16 rows

    const float* kb = k + (size_t)b * N_DIM;
    const float* vb = v + (size_t)b * N_DIM;

    // ---- stage k, v into LDS; track k max/min for stable softmax ----
    float kmx = -3.402823466e38f, kmn = 3.402823466e38f;
#pragma unroll
    for (int it = 0; it < N_DIM / (TPB * 4); ++it) {  // 4 iters of float4
        int j = (it * TPB + tid) * 4;
        v4f kv = *(const v4f*)(kb + j);
        v4f vv = *(const v4f*)(vb + j);
        *(v4f*)(sk + j) = kv;
        *(v4f*)(sv + j) = vv;
        kmx = fmaxf(fmaxf(kmx, kv.x), fmaxf(kv.y, fmaxf(kv.z, kv.w)));
        kmn = fminf(fminf(kmn, kv.x), fminf(kv.y, fminf(kv.z, kv.w)));
    }
    red[tid] = kmx; __syncthreads();
    for (int s = TPB / 2; s > 0; s >>= 1) {
        if (tid < s) red[tid] = fmaxf(red[tid], red[tid + s]);
        __syncthreads();
    }
    if (tid == 0) skmax = red[0];
    __syncthreads();
    red[tid] = kmn; __syncthreads();
    for (int s = TPB / 2; s > 0; s >>= 1) {
        if (tid < s) red[tid] = fminf(red[tid], red[tid + s]);
        __syncthreads();
    }
    if (tid == 0) skmin = red[0];
    __syncthreads();

    // ---- per-row constants: score row is q_i * k / 64 (rank-1) ----
    if (tid < ROWS) {
        float qm = q[(size_t)b * N_DIM + i0 + tid];
        float c1 = qm * (1.44269504088896340736f / 64.0f);  // log2(e)/sqrt(4096)
        sc1[tid] = c1;
        sm2[tid] = fmaxf(c1 * skmax, c1 * skmin);           // row max (log2 domain)
    }
    __syncthreads();

    // ---- pass 1: l_i and acc_i via chained V_WMMA_F32_16X16X4_F32 ----
    // A (16x4 f32): lane L -> M = L%16, K = 2*(L/16) + {vgpr0,vgpr1}
    // B (4x16 f32): lane L -> N = L%16, K = 2*(L/16) + {vgpr0,vgpr1}
    //   column N=0 is all-ones (-> l), column N=1 is v[j] (-> acc), rest 0.
    const int  lane  = tid & 31;
    const int  w     = tid >> 5;
    const int  m     = lane & 15;           // A-matrix row / B-matrix column
    const int  half2 = (lane >> 4) << 1;    // K base: 0 or 2
    const float c1m  = sc1[m];
    const float m2m  = sm2[m];
    const float w0   = (m == 0) ? 1.0f : 0.0f;
    const bool  isn1 = (m == 1);

    v8f c = {};
    const int jw = w * JSTRIPE;
    for (int cb = 0; cb < JSTRIPE; cb += 4) {
        const int jA = jw + cb + half2;
        v2f a, bb;
        a.x = __builtin_amdgcn_exp2f(__builtin_fmaf(c1m, sk[jA],     -m2m));
        a.y = __builtin_amdgcn_exp2f(__builtin_fmaf(c1m, sk[jA + 1], -m2m));
        float v0 = sv[jA], v1 = sv[jA + 1];
        bb.x = isn1 ? v0 : w0;
        bb.y = isn1 ? v1 : w0;
        c = __builtin_amdgcn_wmma_f32_16x16x4_f32(false, a, false, bb,
                                                  (short)0, c, false, false);
    }

    // D layout: vgpr r -> (M=r, N=lane) lanes 0-15; (M=r+8, N=lane-16) lanes 16-31
    {
        const int M0 = (lane < 16) ? 0 : 8;
        const int N0 = (lane < 16) ? lane : lane - 16;
        if (N0 == 0) {
#pragma unroll
            for (int r = 0; r < 8; ++r) lsum[w][M0 + r] = c[r];
        } else if (N0 == 1) {
#pragma unroll
            for (int r = 0; r < 8; ++r) lacc[w][M0 + r] = c[r];
        }
    }
    __syncthreads();

    // ---- cross-wave reduction, write output ----
    if (tid < 16) {
        float l = 0.0f;
#pragma unroll
        for (int ww = 0; ww < WAVES; ++ww) l += lsum[ww][tid];
        slinv[tid] = 1.0f / l;
    } else if (tid < 32) {
        const int mm = tid - 16;
        float a = 0.0f;
#pragma unroll
        for (int ww = 0; ww < WAVES; ++ww) a += lacc[ww][mm];
        sacc[mm] = a;
    }
    __syncthreads();
    if (tid < 16) out[(size_t)b * N_DIM + i0 + tid] = sacc[tid] * slinv[tid];

    // ---- pass 2: stream the normalized distribution (store-BW bound) ----
    // Loop interchange: load each thread's float4 of k ONCE, reuse across all
    // 16 rows; dist is write-once and exceeds L2 -> non-temporal stores.
    const float* drow0 = dist + (size_t)(b * N_DIM + i0) * N_DIM;
#pragma unroll 1
    for (int it = 0; it < N_DIM / (TPB * 4); ++it) {  // 4 j-slices per thread
        const int j = (it * TPB + tid) * 4;
        const v4f kv = *(const v4f*)(sk + j);
        float* p = (float*)drow0 + j;
#pragma unroll
        for (int mm = 0; mm < ROWS; ++mm) {
            const float c1r = sc1[mm];
            const float m2r = sm2[mm];
            const float sc  = slinv[mm];
            v4f e;
            e.x = __builtin_amdgcn_exp2f(__builtin_fmaf(c1r, kv.x, -m2r)) * sc;
            e.y = __builtin_amdgcn_exp2f(__builtin_fmaf(c1r, kv.y, -m2r)) * sc;
            e.z = __builtin_amdgcn_exp2f(__builtin_fmaf(c1r, kv.z, -m2r)) * sc;
            e.w = __builtin_amdgcn_exp2f(__builtin_fmaf(c1r, kv.w, -m2r)) * sc;
            __builtin_nontemporal_store(e, (v4f*)p);
            p += N_DIM;  // next dist row, same j slice
        }
    }
}

extern "C" void kernel_launch(void* const* d_in, const int* in_sizes, int n_in,
                              void* d_out, int out_size, void* d_ws, size_t ws_size,
                              hipStream_t stream) {
    (void)in_sizes; (void)n_in; (void)out_size; (void)d_ws; (void)ws_size;
    // setup_inputs order: query, value, key
    const float* q = (const float*)d_in[0];
    const float* v = (const float*)d_in[1];
    const float* k = (const float*)d_in[2];
    float* out  = (float*)d_out;                          // [16, 4096]
    float* dist = (float*)d_out + (size_t)B_DIM * N_DIM;  // [16, 4096, 4096]

    dim3 grid(B_DIM * (N_DIM / ROWS));  // 4096 blocks, one 16-row tile each
    attn_rank1_softmax<<<grid, TPB, 0, stream>>>(q, v, k, out, dist);
}